// ModifiedModel_24489903522575
// MI455X (gfx1250) — compile-verified
//
#include <hip/hip_runtime.h>
#include <hip/hip_bf16.h>
#include <math.h>

#define B_ROWS 4096
#define D_K    64
#define V_COLS 50257

typedef __attribute__((ext_vector_type(2))) float v2f;
typedef __attribute__((ext_vector_type(8))) float v8f;

__device__ __forceinline__ unsigned f2key(float f) {
    unsigned b = __float_as_uint(f);
    // monotonic order-preserving map float -> uint32
    return (b & 0x80000000u) ? ~b : (b | 0x80000000u);
}

__global__ void init_ws_kernel(unsigned long long* ws) {
    int i = blockIdx.x * blockDim.x + threadIdx.x;
    if (i < B_ROWS) ws[i] = 0ull;  // below any encoded finite/-inf key
}

// Block: 32 rows x 1024 cols. Each of 8 waves: 32 rows x 128 cols
// (2 M-tiles x 8 N-tiles); B registers are reused across both M-tiles,
// halving W refetch traffic from L2.
__global__ __launch_bounds__(256) void logits_wmma_kernel(
        const float* __restrict__ x,      // [B, 64]
        const float* __restrict__ W,      // [V, 64]
        const float* __restrict__ bias,   // [V]
        float* __restrict__ out,          // [B, V]
        unsigned long long* __restrict__ ws) {  // [B] packed (key, ~col)
    const int lane = threadIdx.x & 31;
    const int wave = threadIdx.x >> 5;
    const int lm   = lane & 15;      // position within half-wave
    const int half = lane >> 4;      // 0: rows +0..7 / K pair {4s,4s+1}; 1: rows +8..15 / {4s+2,4s+3}
    const int m0     = blockIdx.y * 32;
    const int n_base = blockIdx.x * 1024 + wave * 128;

    if (n_base >= V_COLS) return;                 // wave-uniform: whole strip out of range
    const bool full = (n_base + 128 <= V_COLS);   // wave-uniform fast-path condition

    // A tiles (x): two 16x64 fp32 tiles. WMMA 16x16x4 A layout per step s:
    //   lanes 0-15 : (M=lane,    K=4s+0..4s+1) -> contiguous float2
    //   lanes 16-31: (M=lane-16, K=4s+2..4s+3) -> contiguous float2
    v2f a[2][16];
#pragma unroll
    for (int mt = 0; mt < 2; ++mt) {
        const float* xr = x + (size_t)(m0 + mt * 16 + lm) * D_K + 2 * half;
#pragma unroll
        for (int s = 0; s < 16; ++s) a[mt][s] = *(const v2f*)(xr + 4 * s);
    }

    float bval[16];
    int   bcol[16];
#pragma unroll
    for (int q = 0; q < 16; ++q) { bval[q] = -INFINITY; bcol[q] = 0x7fffffff; }

    if (full) {
        // ---------- fast path: no predication anywhere ----------
        for (int t = 0; t < 8; ++t) {
            const int col = n_base + t * 16 + lm;             // this lane's output column
            const float* wr = W + (size_t)col * D_K + 2 * half;  // B[k][n] = W[n*64+k]
            v2f b[16];
#pragma unroll
            for (int s = 0; s < 16; ++s) b[s] = *(const v2f*)(wr + 4 * s);
            const float bs = bias[col];
#pragma unroll
            for (int mt = 0; mt < 2; ++mt) {
                v8f c = {};
#pragma unroll
                for (int s = 0; s < 16; ++s)
                    c = __builtin_amdgcn_wmma_f32_16x16x4_f32(
                            false, a[mt][s], false, b[s], (short)0, c, false, false);
#pragma unroll
                for (int r = 0; r < 8; ++r) {
                    const float v   = c[r] + bs;
                    const int   row = m0 + mt * 16 + r + 8 * half;  // D: lanes0-15 -> M=r, lanes16-31 -> M=r+8
                    __builtin_nontemporal_store(v, &out[(size_t)row * V_COLS + col]);
                    // cols strictly increase across t -> strict > keeps lowest index on ties
                    const int q = mt * 8 + r;
                    if (v > bval[q]) { bval[q] = v; bcol[q] = col; }
                }
            }
        }
    } else {
        // ---------- boundary path: clamp loads, predicate stores ----------
        for (int t = 0; t < 8; ++t) {
            const int col = n_base + t * 16 + lm;
            const int ccl = (col < V_COLS) ? col : (V_COLS - 1);
            const float* wr = W + (size_t)ccl * D_K + 2 * half;
            v2f b[16];
#pragma unroll
            for (int s = 0; s < 16; ++s) b[s] = *(const v2f*)(wr + 4 * s);
            const float bs    = bias[ccl];
            const bool  valid = (col < V_COLS);
#pragma unroll
            for (int mt = 0; mt < 2; ++mt) {
                v8f c = {};
#pragma unroll
                for (int s = 0; s < 16; ++s)
                    c = __builtin_amdgcn_wmma_f32_16x16x4_f32(
                            false, a[mt][s], false, b[s], (short)0, c, false, false);
#pragma unroll
                for (int r = 0; r < 8; ++r) {
                    const float v   = c[r] + bs;
                    const int   row = m0 + mt * 16 + r + 8 * half;
                    if (valid) __builtin_nontemporal_store(v, &out[(size_t)row * V_COLS + col]);
                    const int q = mt * 8 + r;
                    if (valid && v > bval[q]) { bval[q] = v; bcol[q] = col; }
                }
            }
        }
    }

    // Per-row reduction across the 16 lanes of each half-wave, then one atomic per row.
#pragma unroll
    for (int q = 0; q < 16; ++q) {
        float v  = bval[q];
        int   cc = bcol[q];
#pragma unroll
        for (int m = 8; m >= 1; m >>= 1) {
            float ov = __shfl_xor(v, m, 32);
            int   oc = __shfl_xor(cc, m, 32);
            if (ov > v || (ov == v && oc < cc)) { v = ov; cc = oc; }
        }
        if (lm == 0 && v != -INFINITY) {
            const int row = m0 + (q >> 3) * 16 + (q & 7) + 8 * half;
            const unsigned long long key =
                ((unsigned long long)f2key(v) << 32) | (unsigned)(~(unsigned)cc);
            atomicMax(&ws[row], key);  // max key wins; equal keys -> smaller col (larger ~col)
        }
    }
}

__global__ void finalize_kernel(const float* __restrict__ rand_vals,
                                const unsigned long long* __restrict__ ws,
                                float* __restrict__ out) {
    int row = blockIdx.x * blockDim.x + threadIdx.x;
    if (row >= B_ROWS) return;
    if (!(rand_vals[row] < 0.5f)) {  // keep_mask == false -> knock out argmax
        unsigned cc = ~(unsigned)(ws[row] & 0xffffffffull);
        out[(size_t)row * V_COLS + cc] = -INFINITY;
    }
}

extern "C" void kernel_launch(void* const* d_in, const int* in_sizes, int n_in,
                              void* d_out, int out_size, void* d_ws, size_t ws_size,
                              hipStream_t stream) {
    const float* x    = (const float*)d_in[0];   // [4096, 64]
    const float* W    = (const float*)d_in[1];   // [50257, 64]
    const float* bias = (const float*)d_in[2];   // [50257]
    const float* rnd  = (const float*)d_in[3];   // [4096]
    float* out = (float*)d_out;                  // [4096, 50257]
    unsigned long long* ws = (unsigned long long*)d_ws;  // 4096 * 8 bytes

    (void)in_sizes; (void)n_in; (void)out_size; (void)ws_size;

    init_ws_kernel<<<(B_ROWS + 255) / 256, 256, 0, stream>>>(ws);

    dim3 grid((V_COLS + 1023) / 1024, B_ROWS / 32);  // (50, 128)
    logits_wmma_kernel<<<grid, 256, 0, stream>>>(x, W, bias, out, ws);

    finalize_kernel<<<(B_ROWS + 255) / 256, 256, 0, stream>>>(rnd, ws, out);
}